// CRZ_88871463288931
// MI455X (gfx1250) — compile-verified
//
#include <hip/hip_runtime.h>
#include <cstdint>

// CRZ diagonal-phase apply: out[i,b] = phase[i] * x[i,b]
//   phase[i] = 1                 if bit11(i)==0
//            = exp(-i*theta/2)   if bit11(i)==1 && bit10(i)==0
//            = exp(+i*theta/2)   if bit11(i)==1 && bit10(i)==1
// x: complex64 (D=4096, BATCH=2048) row-major, interleaved re/im floats.
// Memory-bound: 128 MiB total traffic -> ~5.8 us floor at 23.3 TB/s.
// CDNA5 path: 4-deep async global->LDS pipeline (ASYNCcnt), wave-private
// LDS slots (no barriers), 128-bit loads/stores, scalarized phase select.

namespace {
constexpr int kD      = 4096;                       // 2^12 basis states
constexpr int kBatch  = 2048;
constexpr int kTPB    = 256;                        // 8 wave32 per block
constexpr int kIters  = 8;                          // tiles per block
constexpr int kPipe   = 4;                          // async pipeline depth
constexpr int kVec4   = kD * kBatch / 2;            // float4 count = 4,194,304
constexpr int kBlocks = kVec4 / (kTPB * kIters);    // 2048 blocks, exact cover
}

__device__ __forceinline__ void wait_async_le(int n) {
    // n is a constant after full unroll; switch keeps asm immediates literal.
    switch (n) {
    case 0: asm volatile("s_wait_asynccnt 0x0" ::: "memory"); break;
    case 1: asm volatile("s_wait_asynccnt 0x1" ::: "memory"); break;
    case 2: asm volatile("s_wait_asynccnt 0x2" ::: "memory"); break;
    default: asm volatile("s_wait_asynccnt 0x3" ::: "memory"); break;
    }
}

__global__ __launch_bounds__(kTPB)
void crz_diag_phase_async(const float4* __restrict__ x,
                          const float*  __restrict__ theta_p,
                          float4*       __restrict__ out) {
    // 4 buffers x 256 float4 = 16 KiB LDS (of 320 KiB/WGP)
    __shared__ float4 smem[kPipe * kTPB];

    const int      tid     = threadIdx.x;
    const uint32_t ldsBase = (uint32_t)(uintptr_t)(&smem[0]);   // low 32 bits = LDS offset
    const uint64_t srcBase = (uint64_t)(uintptr_t)x;
    const int      first   = blockIdx.x * (kTPB * kIters);      // first float4 of this block

    // Only 3 distinct phases: compute cos/sin once (uniform per block after
    // the compiler scalarizes the row-bit selects).
    const float th = theta_p[0];
    float sn, cs;
    __sincosf(0.5f * th, &sn, &cs);

    // Prologue: fill the pipeline with kPipe-1 tiles.
#pragma unroll
    for (int t = 0; t < kPipe - 1; ++t) {
        const uint32_t lds = ldsBase + (uint32_t)(t * kTPB + tid) * 16u;
        const uint32_t off = (uint32_t)(first + t * kTPB + tid) * 16u;  // < 2^27
        asm volatile("global_load_async_to_lds_b128 %0, %1, %2"
                     :: "v"(lds), "v"(off), "s"(srcBase) : "memory");
    }

#pragma unroll
    for (int t = 0; t < kIters; ++t) {
        const int buf = t % kPipe;
        if (t + kPipe - 1 < kIters) {
            // Prefetch tile t+kPipe-1 into the buffer freed at iter t-1.
            const int pbuf = (t + kPipe - 1) % kPipe;
            const uint32_t lds = ldsBase + (uint32_t)(pbuf * kTPB + tid) * 16u;
            const uint32_t off = (uint32_t)(first + (t + kPipe - 1) * kTPB + tid) * 16u;
            asm volatile("global_load_async_to_lds_b128 %0, %1, %2"
                         :: "v"(lds), "v"(off), "s"(srcBase) : "memory");
            wait_async_le(kPipe - 1);       // in-order completion: tile t landed
        } else {
            wait_async_le(kIters - 1 - t);  // drain phase
        }

        // Each lane reads exactly the LDS slot its own async load targeted
        // (ASYNCcnt is per-wave -> no barriers needed).
        const float4   v   = smem[buf * kTPB + tid];
        const uint32_t g   = (uint32_t)(first + t * kTPB + tid); // float4 index
        const uint32_t row = g >> 10;                            // 1024 float4 per row
        const uint32_t loc = (row >> 11) & 1u;                   // control digit
        const uint32_t k   = (row >> 10) & 1u;                   // target digit
        const float cr = loc ? cs : 1.0f;
        const float ci = loc ? (k ? sn : -sn) : 0.0f;

        // (cr + i*ci) * (re + i*im), two complex values per float4
        float4 o;
        o.x = fmaf(cr, v.x, -ci * v.y);
        o.y = fmaf(cr, v.y,  ci * v.x);
        o.z = fmaf(cr, v.z, -ci * v.w);
        o.w = fmaf(cr, v.w,  ci * v.z);
        out[g] = o;
    }
}

extern "C" void kernel_launch(void* const* d_in, const int* in_sizes, int n_in,
                              void* d_out, int out_size, void* d_ws, size_t ws_size,
                              hipStream_t stream) {
    const float4* x     = (const float4*)d_in[0];   // complex64 as float pairs
    const float*  theta = (const float*)d_in[1];    // single float
    float4*       out   = (float4*)d_out;
    crz_diag_phase_async<<<kBlocks, kTPB, 0, stream>>>(x, theta, out);
}